// MessagePassing_41480794145043
// MI455X (gfx1250) — compile-verified
//
#include <hip/hip_runtime.h>

// ---------------------------------------------------------------------------
// Graph message passing: scatter-mean over E=12.8M edges into N=200K nodes
// (D=10), then out = relu(agg @ (W+B)^T).
//
// MI455X plan:
//  * Edge phase: HBM cost is only the edge-index streams (~102 MB -> ~4.4 us
//    at 23.3 TB/s). node_features (8 MB) + padded accumulators (10.4 MB) are
//    L2-resident (192 MB), so the 128M gathers and 141M no-return
//    global_atomic_add_f32 are L2 traffic. NT loads keep the streamed indices
//    from evicting the hot node data.
//  * Node phase: V_WMMA_F32_16X16X4_F32 (f32 => exact precision), one wave
//    per 16-node tile, K=10 padded to 12 = three K=4 WMMA steps. Padding is
//    structural (stride-12 accumulator rows, zeroed up-front; LDS-staged
//    16x12 column-major (W+B)^T), so every fragment load is an unconditional
//    aligned 64-bit load. Stores take a wave-uniform full-tile fast path:
//    EXEC set once for l<10, then 8 global_store_b32 off one base register.
// ---------------------------------------------------------------------------

typedef __attribute__((ext_vector_type(2))) float v2f;
typedef __attribute__((ext_vector_type(8))) float v8f;
typedef __attribute__((ext_vector_type(2))) int   v2i;

#define FD 10   // feature dim
#define KP 12   // padded K stride (3 x K=4 WMMA steps)

// ---- zero the workspace accumulators ---------------------------------------
__global__ void mp_zero_kernel(float* __restrict__ ws, int n) {
    int i = blockIdx.x * blockDim.x + threadIdx.x;
    if (i < n) ws[i] = 0.0f;
}

// ---- edge scatter: agg[dst] += nf[src]; cnt[dst] += 1 ----------------------
__device__ __forceinline__ void scatter_one(const float* __restrict__ nf,
                                            float* __restrict__ agg,
                                            float* __restrict__ cnt,
                                            int s, int d) {
    const float* fs = nf + (size_t)s * FD;   // row stride 40 B -> 8 B aligned
    float*       ad = agg + (size_t)d * KP;  // padded row, 48 B stride
#pragma unroll
    for (int k = 0; k < FD; k += 2) {
        v2f v = *(const v2f*)(fs + k);       // 8 B aligned (k even)
        unsafeAtomicAdd(ad + k,     v[0]);   // global_atomic_add_f32, no-return
        unsafeAtomicAdd(ad + k + 1, v[1]);
    }
    unsafeAtomicAdd(cnt + d, 1.0f);
}

__global__ void mp_edge_kernel(const float* __restrict__ nf,
                               const int* __restrict__ esrc,
                               const int* __restrict__ edst,
                               float* __restrict__ agg,
                               float* __restrict__ cnt,
                               int E) {
    int e = (blockIdx.x * blockDim.x + threadIdx.x) * 2;
    if (e >= E) return;

    // One lane per wave prefetches the index streams ahead (global_prefetch_b8).
    if ((threadIdx.x & 31) == 0) {
        __builtin_prefetch(esrc + e + 8192, 0, 0);
        __builtin_prefetch(edst + e + 8192, 0, 0);
    }

    if (e + 1 < E) {
        v2i s2 = __builtin_nontemporal_load((const v2i*)(esrc + e)); // 8B NT load
        v2i d2 = __builtin_nontemporal_load((const v2i*)(edst + e));
        scatter_one(nf, agg, cnt, s2[0], d2[0]);
        scatter_one(nf, agg, cnt, s2[1], d2[1]);
    } else {
        int s = __builtin_nontemporal_load(esrc + e);
        int d = __builtin_nontemporal_load(edst + e);
        scatter_one(nf, agg, cnt, s, d);
    }
}

// ---- node transform: out = relu((agg/cnt) @ (W+B)^T) via WMMA f32 16x16x4 --
// f32 16x4 A layout (ISA 7.12.2): lane l and lane l+16 both hold row M=l;
// low half holds K = {k0, k0+1}, high half K = {k0+2, k0+3}. B (4x16)
// mirrors with N striped across lanes. C/D: VGPR r = row r (lanes 0-15) and
// row r+8 (lanes 16-31), N = lane%16.
__global__ void mp_node_kernel(const float* __restrict__ agg,   // [N, KP]
                               const float* __restrict__ cnt,   // [N]
                               const float* __restrict__ W,
                               const float* __restrict__ Bm,
                               float* __restrict__ out,         // [N, FD]
                               int N) {
    // Column-major padded (W+B)^T: MtLDS[col*KP + k] = W[col][k] + B[col][k]
    __shared__ float MtLDS[16 * KP];
    int tid = threadIdx.x;
    if (tid < 16 * KP) {
        int col = tid / KP, k = tid % KP;
        float v = 0.0f;
        if (col < FD && k < FD) v = W[col * FD + k] + Bm[col * FD + k];
        MtLDS[tid] = v;
    }
    __syncthreads();   // before any wave can exit

    int wave   = (int)((blockIdx.x * blockDim.x + tid) >> 5);
    int nTiles = (N + 15) >> 4;
    if (wave >= nTiles) return;              // wave-uniform: EXEC stays all-1s

    int lane = tid & 31;
    int half = lane >> 4;                    // 0: K pair {0,1}, 1: K pair {2,3}
    int l    = lane & 15;                    // row for A, col for B

    int node  = wave * 16 + l;
    int nodeC = node < N ? node : N - 1;     // clamp; stores are masked below

    float inv = 1.0f / fmaxf(cnt[nodeC], 1.0f);

    const float* arow = agg   + (size_t)nodeC * KP + half * 2;  // global, 8B aligned
    const float* bcol = MtLDS + l * KP + half * 2;              // LDS, 8B aligned

    v8f acc = {};
#pragma unroll
    for (int kb = 0; kb < 3; ++kb) {
        v2f ap = *(const v2f*)(arow + kb * 4);   // global_load_b64 (pad cols = 0)
        v2f bp = *(const v2f*)(bcol + kb * 4);   // ds_load_b64     (pad rows = 0)
        v2f a;  a[0] = ap[0] * inv;  a[1] = ap[1] * inv;   // mean on the fly
        // 8 args: (neg_a, A, neg_b, B, c_mod, C, reuse_a, reuse_b)
        acc = __builtin_amdgcn_wmma_f32_16x16x4_f32(
            false, a, false, bp, (short)0, acc, false, false);
    }

    // ---- epilogue: rows for this lane are wave*16 + half*8 + r, col = l ----
    float* orow = out + (size_t)(wave * 16 + half * 8) * FD + l;
    bool fullTile = (wave * 16 + 16) <= N;   // wave-uniform branch

    if (fullTile) {
        if (l < FD) {                        // single EXEC mask for all 8 stores
#pragma unroll
            for (int r = 0; r < 8; ++r) {
                float x = acc[r];
                orow[r * FD] = x > 0.0f ? x : 0.0f;      // ReLU
            }
        }
    } else {
#pragma unroll
        for (int r = 0; r < 8; ++r) {
            int row = wave * 16 + half * 8 + r;
            if (l < FD && row < N) {
                float x = acc[r];
                orow[r * FD] = x > 0.0f ? x : 0.0f;
            }
        }
    }
}

// ---------------------------------------------------------------------------
extern "C" void kernel_launch(void* const* d_in, const int* in_sizes, int n_in,
                              void* d_out, int out_size, void* d_ws, size_t ws_size,
                              hipStream_t stream) {
    const float* nf   = (const float*)d_in[0];  // [N, 10]
    const float* W    = (const float*)d_in[1];  // [10, 10]
    const float* Bm   = (const float*)d_in[2];  // [10, 10]
    const int*   esrc = (const int*)d_in[3];    // [E]
    const int*   edst = (const int*)d_in[4];    // [E]
    float*       out  = (float*)d_out;          // [N, 10]

    const int N = in_sizes[0] / FD;
    const int E = in_sizes[3];

    float* agg = (float*)d_ws;                  // N*KP floats (stride-12 rows)
    float* cnt = agg + (size_t)N * KP;          // N floats

    // 1) zero accumulators (every call: graph replay reuses ws)
    int zn = N * KP + N;
    mp_zero_kernel<<<(zn + 255) / 256, 256, 0, stream>>>((float*)d_ws, zn);

    // 2) edge scatter: 2 edges/thread, atomics into L2-resident accumulators
    int ethreads = (E + 1) / 2;
    mp_edge_kernel<<<(ethreads + 255) / 256, 256, 0, stream>>>(nf, esrc, edst,
                                                               agg, cnt, E);

    // 3) node transform with WMMA (one wave per 16-node tile)
    int nTiles  = (N + 15) / 16;
    int threads = nTiles * 32;
    mp_node_kernel<<<(threads + 255) / 256, 256, 0, stream>>>(agg, cnt, W, Bm,
                                                              out, N);
}